// ConvCaps_5849745457705
// MI455X (gfx1250) — compile-verified
//
#include <hip/hip_runtime.h>
#include <math.h>

// ---- problem constants (from reference) ----
#define BCAPS   32
#define CCAPS   32
#define PP      4
#define PSIZE   16
#define BK      288          // K*K*B input capsules per position
#define NPOS    392          // 8 * 7 * 7
#define XCH     544          // B*(PSIZE+1)
#define EMITERS 3
#define EPSF    1e-8f
#define LN_2PI  1.8378770664093453f
#define OUTC    (CCAPS*PSIZE + CCAPS)   // 544

typedef float v2f __attribute__((ext_vector_type(2)));
typedef float v8f __attribute__((ext_vector_type(8)));

// Decode flat index of the reference's (8,3,3,7,7,512) pose-patch array into
// a flat index of x (8,16,16,544). Replicates _add_patches + reshape exactly.
__device__ __forceinline__ int decode_pose_idx(int l) {
    int cc = l % 512; l /= 512;
    int ow = l % 7;   l /= 7;
    int oh = l % 7;   l /= 7;
    int k2 = l % 3;   l /= 3;
    int k1 = l % 3;   l /= 3;
    int bi = l;
    int h = oh * 2 + k1, w = ow * 2 + k2;
    return ((bi * 16 + h) * 16 + w) * XCH + cc;
}

// Same for the (8,3,3,7,7,32) activation slice (channels 512..543 of x).
__device__ __forceinline__ int decode_act_idx(int l) {
    int c32 = l % 32; l /= 32;
    int ow  = l % 7;  l /= 7;
    int oh  = l % 7;  l /= 7;
    int k2  = l % 3;  l /= 3;
    int k1  = l % 3;  l /= 3;
    int bi  = l;
    int h = oh * 2 + k1, w = ow * 2 + k2;
    return ((bi * 16 + h) * 16 + w) * XCH + 512 + c32;
}

// ---------------------------------------------------------------------------
// Kernel 1: votes via V_WMMA_F32_16X16X4_F32.
// One wave per (group-of-4 positions, input capsule b).
//   A[16x4]  : row = pos*4 + p, col = q   -> X[n,b,p,q]  (pose matrix rows)
//   B[4x16]  : row = q, col = c_local*4+r -> W[b, t*4+c_local, q, r]
//   D[16x16] : V[n0+pos, b, c, p*4+r], 8 tiles t cover all 32 c.
// Block = 256 threads = 8 waves = 8 consecutive b's for one group.
// ---------------------------------------------------------------------------
__global__ void conv_caps_votes_kernel(const float* __restrict__ x,
                                       const float* __restrict__ wts,
                                       float* __restrict__ v_ws,
                                       int pos_base) {
    const int tid   = threadIdx.x;
    const int lane  = tid & 31;
    const int wave  = tid >> 5;
    const int gloc  = blockIdx.x / 36;          // group within this chunk
    const int b     = (blockIdx.x % 36) * 8 + wave;
    const int n0    = pos_base + gloc * 4;      // global position base
    const int nloc0 = gloc * 4;                 // chunk-local position base

    const int half = lane >> 4;                 // 0: K=0,1  1: K=2,3
    const int lrow = lane & 15;

    // --- A fragment: lane holds A[lrow][half*2 + j], j = 0,1 ---
    const int pos = lrow >> 2;
    const int p   = lrow & 3;
    v2f a;
    {
        const int base_l = ((n0 + pos) * BK + b) * PSIZE + p * 4 + half * 2;
        a[0] = x[decode_pose_idx(base_l + 0)];
        a[1] = x[decode_pose_idx(base_l + 1)];
    }

    const int c_local = lrow >> 2;              // B/D column -> (c_local, r)
    const int r       = lrow & 3;

#pragma unroll
    for (int t = 0; t < 8; ++t) {
        const int c = t * 4 + c_local;
        // --- B fragment: lane holds B[half*2 + j][lrow] = W[b][c][q][r] ---
        v2f bf;
        const float* wp = wts + (((b * CCAPS + c) * PP + half * 2) * PP) + r;
        bf[0] = wp[0];
        bf[1] = wp[PP];
        v8f acc = {};
        acc = __builtin_amdgcn_wmma_f32_16x16x4_f32(
                  /*neg_a=*/false, a, /*neg_b=*/false, bf,
                  /*c_mod=*/(short)0, acc, /*reuse_a=*/false, /*reuse_b=*/false);
        // --- D: VGPR vv -> row = vv + 8*half, col = lrow ---
#pragma unroll
        for (int vv = 0; vv < 8; ++vv) {
            const int row  = vv + half * 8;
            const int dpos = row >> 2, dp = row & 3;
            const int nloc = nloc0 + dpos;
            v_ws[((nloc * BK + b) * CCAPS + c) * PSIZE + dp * 4 + r] = acc[vv];
        }
    }
}

// ---------------------------------------------------------------------------
// Kernel 2: EM routing, one 256-thread workgroup per position.
// r (288x32) lives in LDS (36 KB) and is reused for ln_ap in the E-step.
// Votes are read from the (L2-resident) chunk workspace.
// ---------------------------------------------------------------------------
__global__ void conv_caps_routing_kernel(const float* __restrict__ x,
                                         const float* __restrict__ v_ws,
                                         const float* __restrict__ beta_u,
                                         const float* __restrict__ beta_a,
                                         float* __restrict__ out,
                                         int pos_base) {
    __shared__ float r_sm[BK * CCAPS];          // 36 KB; also holds ln_ap
    __shared__ float a_in_sm[BK];
    __shared__ float r_sum_sm[CCAPS];
    __shared__ float mu_sm[CCAPS * PSIZE];
    __shared__ float sig_sm[CCAPS * PSIZE];
    __shared__ float a_out_sm[CCAPS];

    const int tid  = threadIdx.x;
    const int nloc = blockIdx.x;
    const int n    = pos_base + nloc;
    const float* v = v_ws + (size_t)nloc * BK * CCAPS * PSIZE;

    for (int b = tid; b < BK; b += blockDim.x)
        a_in_sm[b] = x[decode_act_idx(n * BK + b)];
    for (int i = tid; i < BK * CCAPS; i += blockDim.x)
        r_sm[i] = 1.0f / (float)CCAPS;
    __syncthreads();

    for (int it = 0; it < EMITERS; ++it) {
        const float inv_temp = 0.01f * (1.0f - powf(0.95f, (float)(it + 1)));

        // (a) r *= a_in; normalize over c (per b)
        for (int b = tid; b < BK; b += blockDim.x) {
            float* rb = &r_sm[b * CCAPS];
            const float ai = a_in_sm[b];
            float s = 0.0f;
            for (int c = 0; c < CCAPS; ++c) { rb[c] *= ai; s += rb[c]; }
            const float inv = 1.0f / (s + EPSF);
            for (int c = 0; c < CCAPS; ++c) rb[c] *= inv;
        }
        __syncthreads();

        // (b) r_sum[c] = sum_b r
        if (tid < CCAPS) {
            float s = 0.0f;
            for (int b = 0; b < BK; ++b) s += r_sm[b * CCAPS + tid];
            r_sum_sm[tid] = s;
        }
        __syncthreads();

        // (c) mu / sigma per (c,i); sigma = E[v^2] - mu^2*(2 - S) + eps
        for (int ci = tid; ci < CCAPS * PSIZE; ci += blockDim.x) {
            const int c = ci >> 4, i = ci & 15;
            float s1 = 0.0f, s2 = 0.0f;
            for (int b = 0; b < BK; ++b) {
                const float rv = r_sm[b * CCAPS + c];
                const float vv = v[(b * CCAPS + c) * PSIZE + i];
                s1 += rv * vv;
                s2 += rv * vv * vv;
            }
            const float rs  = r_sum_sm[c];
            const float rse = rs + EPSF;
            const float mu  = s1 / rse;
            const float S   = rs / rse;
            mu_sm[ci]  = mu;
            sig_sm[ci] = s2 / rse - mu * mu * (2.0f - S) + EPSF;
        }
        __syncthreads();

        // (d) a_out[c] = sigmoid(inv_temp * (beta_a - cost))
        if (tid < CCAPS) {
            float cost = 0.0f;
            const float bu = beta_u[tid];
            for (int i = 0; i < PSIZE; ++i)
                cost += bu + 0.5f * logf(sig_sm[tid * PSIZE + i]);
            cost *= r_sum_sm[tid];
            const float z = inv_temp * (beta_a[tid] - cost);
            a_out_sm[tid] = 1.0f / (1.0f + expf(-z));
        }
        __syncthreads();

        if (it < EMITERS - 1) {
            // E-step: ln_ap into r_sm, then per-b softmax over c
            for (int bc = tid; bc < BK * CCAPS; bc += blockDim.x) {
                const int b = bc >> 5, c = bc & 31;
                const float* vp = &v[(b * CCAPS + c) * PSIZE];
                float s = 0.0f;
                for (int i = 0; i < PSIZE; ++i) {
                    const float d  = vp[i] - mu_sm[c * PSIZE + i];
                    const float sg = sig_sm[c * PSIZE + i];
                    s += -d * d / (2.0f * sg) - 0.5f * logf(sg);
                }
                r_sm[bc] = s - 0.5f * LN_2PI * (float)PSIZE + logf(a_out_sm[c]);
            }
            __syncthreads();
            for (int b = tid; b < BK; b += blockDim.x) {
                float* rb = &r_sm[b * CCAPS];
                float m = rb[0];
                for (int c = 1; c < CCAPS; ++c) m = fmaxf(m, rb[c]);
                float s = 0.0f;
                for (int c = 0; c < CCAPS; ++c) { const float e = expf(rb[c] - m); rb[c] = e; s += e; }
                const float inv = 1.0f / s;
                for (int c = 0; c < CCAPS; ++c) rb[c] *= inv;
            }
            __syncthreads();
        }
    }

    // outputs: (8,7,7, C*16 | C)
    float* o = out + (size_t)n * OUTC;
    for (int ci = tid; ci < CCAPS * PSIZE; ci += blockDim.x) o[ci] = mu_sm[ci];
    if (tid < CCAPS) o[CCAPS * PSIZE + tid] = a_out_sm[tid];
}

// ---------------------------------------------------------------------------
extern "C" void kernel_launch(void* const* d_in, const int* in_sizes, int n_in,
                              void* d_out, int out_size, void* d_ws, size_t ws_size,
                              hipStream_t stream) {
    (void)in_sizes; (void)n_in; (void)out_size;
    const float* x  = (const float*)d_in[0];
    const float* w  = (const float*)d_in[1];   // (1,288,32,4,4)
    const float* bu = (const float*)d_in[2];
    const float* ba = (const float*)d_in[3];
    float* out = (float*)d_out;
    float* vws = (float*)d_ws;

    // Bytes of votes per group of 4 positions; chunk so votes stay L2-resident
    // and within ws_size. 98 groups total.
    const size_t bytes_per_group = (size_t)4 * BK * CCAPS * PSIZE * sizeof(float);
    size_t max_groups = ws_size / bytes_per_group;
    if (max_groups < 1) max_groups = 1;           // hope ws >= 2.36 MB
    if (max_groups > 98) max_groups = 98;
    const int G = (int)max_groups;

    for (int g0 = 0; g0 < 98; g0 += G) {
        const int groups = (98 - g0 < G) ? (98 - g0) : G;
        const int pos_base = g0 * 4;
        conv_caps_votes_kernel<<<groups * 36, 256, 0, stream>>>(x, w, vws, pos_base);
        conv_caps_routing_kernel<<<groups * 4, 256, 0, stream>>>(x, vws, bu, ba, out, pos_base);
    }
}